// GCN_13073880449099
// MI455X (gfx1250) — compile-verified
//
#include <hip/hip_runtime.h>

typedef __attribute__((ext_vector_type(2))) float v2f;
typedef __attribute__((ext_vector_type(8))) float v8f;

#define N_NODES_C 100000
#define N_EDGES_C 1600000
#define IN_F 256
#define OUT_F 64
#define ROWS_PER_BLOCK 32
#define CHUNK_K 64

// ---------------------------------------------------------------------------
// CDNA5 async global->LDS helpers (ASYNCcnt-tracked, no VGPR round-trip)
// ---------------------------------------------------------------------------
__device__ __forceinline__ void async_load_b128(unsigned lds_off, const float* g)
{
    asm volatile("global_load_async_to_lds_b128 %0, %1, off"
                 :: "v"(lds_off), "v"((unsigned long long)(size_t)g)
                 : "memory");
}
__device__ __forceinline__ void wait_async0()
{
    asm volatile("s_wait_asynccnt 0x0" ::: "memory");
}

// ---------------------------------------------------------------------------
// Kernel 1: support = features @ W   via V_WMMA_F32_16X16X4_F32
// Block: 256 threads = 8 waves; wave (mt,nt) owns a 16x16 tile of a 32x64
// block. K is processed in 64-wide chunks, double-buffered in LDS with
// async global->LDS copies overlapping the WMMA stream.
// ---------------------------------------------------------------------------
__global__ __launch_bounds__(256) void gcn_gemm_wmma(
    const float* __restrict__ A,   // [N_NODES, IN_F]
    const float* __restrict__ W,   // [IN_F, OUT_F]
    float* __restrict__ S)         // [N_NODES, OUT_F]
{
    __shared__ float ldsA[2][ROWS_PER_BLOCK * CHUNK_K];  // 2 x 8 KB
    __shared__ float ldsW[2][CHUNK_K * OUT_F];           // 2 x 16 KB, row-major [k][n]

    const int tid  = threadIdx.x;
    const int lane = tid & 31;
    const int wave = tid >> 5;
    const int mt   = wave >> 2;     // 0..1  (M tile)
    const int nt   = wave & 3;      // 0..3  (N tile)
    const int half = lane >> 4;     // 0 or 1
    const int l15  = lane & 15;
    const int rowBase = blockIdx.x * ROWS_PER_BLOCK;

    v8f c = {};

    // ---- async staging of one K-chunk into LDS buffer `buf` ----
    auto stage_chunk = [&](int buf, int kc) {
        // A chunk: 32 rows x 64 cols = 512 float4 -> 2 b128 per thread
        #pragma unroll
        for (int p = 0; p < 2; ++p) {
            int idx = p * 256 + tid;
            int r   = idx >> 4;
            int cv  = idx & 15;
            const float* g = A + (size_t)(rowBase + r) * IN_F + kc + cv * 4;
            unsigned l = (unsigned)(size_t)&ldsA[buf][r * CHUNK_K + cv * 4];
            async_load_b128(l, g);
        }
        // W chunk: rows kc..kc+63 are contiguous (64*64 floats) -> flat copy
        #pragma unroll
        for (int p = 0; p < 4; ++p) {
            int idx = p * 256 + tid;
            const float* g = W + (size_t)kc * OUT_F + idx * 4;
            unsigned l = (unsigned)(size_t)&ldsW[buf][idx * 4];
            async_load_b128(l, g);
        }
    };

    // prologue: fill buffer 0
    stage_chunk(0, 0);
    wait_async0();
    __syncthreads();

    #pragma unroll
    for (int ch = 0; ch < IN_F / CHUNK_K; ++ch) {
        const int cur = ch & 1;
        // prefetch next chunk into the other buffer while we compute
        if (ch + 1 < IN_F / CHUNK_K)
            stage_chunk(cur ^ 1, (ch + 1) * CHUNK_K);

        // 16 WMMA steps over this chunk
        const float* aB = &ldsA[cur][(mt * 16 + l15) * CHUNK_K];
        const float* bB = &ldsW[cur][nt * 16 + l15];
        #pragma unroll
        for (int kk = 0; kk < CHUNK_K; kk += 4) {
            int ko = kk + 2 * half;             // per-lane K pair per ISA layout
            v2f a, b;
            a.x = aB[ko];
            a.y = aB[ko + 1];
            b.x = bB[(ko    ) * OUT_F];
            b.y = bB[(ko + 1) * OUT_F];
            c = __builtin_amdgcn_wmma_f32_16x16x4_f32(
                    false, a, false, b, (short)0, c, false, false);
        }

        if (ch + 1 < IN_F / CHUNK_K) {
            wait_async0();      // our async fills for the next buffer landed
            __syncthreads();    // everyone's fills landed; everyone done reading
        }
    }

    // --- store 16x16 tile: VGPR r -> M = r + 8*half, N = l15 ---
    const int N = nt * 16 + l15;
    const int Mbase = rowBase + mt * 16 + 8 * half;
    #pragma unroll
    for (int r = 0; r < 8; ++r) {
        S[(size_t)(Mbase + r) * OUT_F + N] = c[r];
    }
}

// ---------------------------------------------------------------------------
// Kernel 2: zero the accumulator (d_out)
// ---------------------------------------------------------------------------
__global__ __launch_bounds__(256) void gcn_zero(float4* __restrict__ out, int n4)
{
    int i = blockIdx.x * blockDim.x + threadIdx.x;
    if (i < n4) out[i] = make_float4(0.f, 0.f, 0.f, 0.f);
}

// ---------------------------------------------------------------------------
// Kernel 3: edge-parallel SpMM scatter-add. 16 threads/edge, float4 gather,
// 4 hw fp32 atomics per thread. support + out both L2-resident (25.6 MB each).
// ---------------------------------------------------------------------------
__global__ __launch_bounds__(256) void gcn_spmm_scatter(
    const float* __restrict__ support,
    const int* __restrict__ adj_row,
    const int* __restrict__ adj_col,
    const float* __restrict__ adj_val,
    float* __restrict__ out,
    int nEdges)
{
    int tid = blockIdx.x * blockDim.x + threadIdx.x;
    int e = tid >> 4;
    if (e >= nEdges) return;
    int f = (tid & 15) * 4;

    int cIdx = adj_col[e];
    int rIdx = adj_row[e];
    float v  = adj_val[e];

    float4 s = *(const float4*)(support + (size_t)cIdx * OUT_F + f);
    float* dst = out + (size_t)rIdx * OUT_F + f;

    __hip_atomic_fetch_add(dst + 0, s.x * v, __ATOMIC_RELAXED, __HIP_MEMORY_SCOPE_AGENT);
    __hip_atomic_fetch_add(dst + 1, s.y * v, __ATOMIC_RELAXED, __HIP_MEMORY_SCOPE_AGENT);
    __hip_atomic_fetch_add(dst + 2, s.z * v, __ATOMIC_RELAXED, __HIP_MEMORY_SCOPE_AGENT);
    __hip_atomic_fetch_add(dst + 3, s.w * v, __ATOMIC_RELAXED, __HIP_MEMORY_SCOPE_AGENT);
}

// ---------------------------------------------------------------------------
// Kernel 4: in-place ReLU
// ---------------------------------------------------------------------------
__global__ __launch_bounds__(256) void gcn_relu(float4* __restrict__ out, int n4)
{
    int i = blockIdx.x * blockDim.x + threadIdx.x;
    if (i < n4) {
        float4 v = out[i];
        v.x = fmaxf(v.x, 0.f);
        v.y = fmaxf(v.y, 0.f);
        v.z = fmaxf(v.z, 0.f);
        v.w = fmaxf(v.w, 0.f);
        out[i] = v;
    }
}

extern "C" void kernel_launch(void* const* d_in, const int* in_sizes, int n_in,
                              void* d_out, int out_size, void* d_ws, size_t ws_size,
                              hipStream_t stream)
{
    const float* features = (const float*)d_in[0];
    const float* weight   = (const float*)d_in[1];
    const int*   adj_row  = (const int*)d_in[2];
    const int*   adj_col  = (const int*)d_in[3];
    const float* adj_val  = (const float*)d_in[4];
    float* out     = (float*)d_out;
    float* support = (float*)d_ws;           // 100000*64 floats = 25.6 MB

    const int n4 = (N_NODES_C * OUT_F) / 4;  // 1.6M float4

    // zero the accumulator (independent of GEMM; stream-ordered before scatter)
    gcn_zero<<<(n4 + 255) / 256, 256, 0, stream>>>((float4*)out, n4);

    // dense GEMM via WMMA: 3125 blocks of 32 rows
    gcn_gemm_wmma<<<N_NODES_C / ROWS_PER_BLOCK, 256, 0, stream>>>(features, weight, support);

    // edge-parallel scatter-add: 16 threads per edge
    long long totalThreads = (long long)N_EDGES_C * 16;
    int blocks = (int)((totalThreads + 255) / 256);
    gcn_spmm_scatter<<<blocks, 256, 0, stream>>>(support, adj_row, adj_col, adj_val,
                                                 out, N_EDGES_C);

    // fused activation pass
    gcn_relu<<<(n4 + 255) / 256, 256, 0, stream>>>((float4*)out, n4);
}